// GraphEncoderGCN_6536940224753
// MI455X (gfx1250) — compile-verified
//
#include <hip/hip_runtime.h>
#include <hip/hip_bf16.h>

typedef float v2f __attribute__((ext_vector_type(2)));
typedef float v8f __attribute__((ext_vector_type(8)));

constexpr int C = 256;           // c_in == c_hid == 256
constexpr int NUM_GRAPHS = 128;

// ---------------- utility ----------------
__global__ void zero_kernel(float* __restrict__ p, size_t n) {
  size_t i = (size_t)blockIdx.x * blockDim.x + threadIdx.x;
  size_t stride = (size_t)gridDim.x * blockDim.x;
  for (; i < n; i += stride) p[i] = 0.0f;
}

// deg[dst] += 1 for every edge, plus self-loops
__global__ void degree_kernel(const int* __restrict__ ei, float* __restrict__ deg,
                              int E, int n) {
  int i = blockIdx.x * blockDim.x + threadIdx.x;
  int total = E + n;
  if (i >= total) return;
  int dst = (i < E) ? ei[E + i] : (i - E);
  atomicAdd(&deg[dst], 1.0f);
}

// deg -> deg^{-1/2} in place
__global__ void rsqrt_kernel(float* __restrict__ deg, int n) {
  int i = blockIdx.x * blockDim.x + threadIdx.x;
  if (i >= n) return;
  float d = deg[i];
  deg[i] = (d > 0.0f) ? rsqrtf(d) : 0.0f;
}

// ---------------- fp32 WMMA GEMM:  D[n x C] = A[n x C] * W[C x C] ----------------
// block = 256 threads = 8 waves; block tile = 128 rows x 64 cols; wave tile = 16 x 64.
// A-frag (16x4 f32, 2 VGPR): lane L holds A[L%16][k + 2*(L/16) + {0,1}]
// B-frag (4x16 f32, 2 VGPR): lane L holds B[k + 2*(L/16) + {0,1}][L%16]
// C/D (16x16 f32, 8 VGPR):   vgpr r, lane L -> D[r + 8*(L/16)][L%16]
__global__ void __launch_bounds__(256)
gemm_kernel(const float* __restrict__ A, const float* __restrict__ W,
            float* __restrict__ D, int n) {
  const int wave = threadIdx.x >> 5;
  const int lane = threadIdx.x & 31;
  const int m    = lane & 15;
  const int kh   = (lane >> 4) << 1;     // 0 or 2
  const int row0 = blockIdx.x * 128 + wave * 16;
  const int col0 = blockIdx.y * 64;

  int arow = row0 + m;
  if (arow >= n) arow = n - 1;           // safe clamped read; masked at store
  const float* Aptr = A + (size_t)arow * C;

  v8f acc0 = {}, acc1 = {}, acc2 = {}, acc3 = {};
  #pragma unroll 4
  for (int k = 0; k < C; k += 4) {
    v2f a = *(const v2f*)(Aptr + k + kh);
    const float* w0 = W + (size_t)(k + kh) * C + col0 + m;
    const float* w1 = w0 + C;
    v2f b0 = { w0[0],  w1[0]  };
    v2f b1 = { w0[16], w1[16] };
    v2f b2 = { w0[32], w1[32] };
    v2f b3 = { w0[48], w1[48] };
    acc0 = __builtin_amdgcn_wmma_f32_16x16x4_f32(false, a, false, b0, (short)0, acc0, false, false);
    acc1 = __builtin_amdgcn_wmma_f32_16x16x4_f32(false, a, false, b1, (short)0, acc1, false, false);
    acc2 = __builtin_amdgcn_wmma_f32_16x16x4_f32(false, a, false, b2, (short)0, acc2, false, false);
    acc3 = __builtin_amdgcn_wmma_f32_16x16x4_f32(false, a, false, b3, (short)0, acc3, false, false);
  }

  const int rbase = row0 + ((lane >> 4) << 3);
  #pragma unroll
  for (int r = 0; r < 8; ++r) {
    int rr = rbase + r;
    if (rr < n) {
      float* drow = D + (size_t)rr * C + col0 + m;
      drow[0]  = acc0[r];
      drow[16] = acc1[r];
      drow[32] = acc2[r];
      drow[48] = acc3[r];
    }
  }
}

// ---------------- edge scatter: out[dst] += dis[src]*dis[dst] * feat[src] ----------------
// one wave per (virtual) edge; lane L handles channels [8L, 8L+8)
__global__ void __launch_bounds__(256)
scatter_kernel(const int* __restrict__ ei, const float* __restrict__ dis,
               const float* __restrict__ feat, float* __restrict__ out,
               int E, int n) {
  const int wave = threadIdx.x >> 5;
  const int lane = threadIdx.x & 31;
  const int e = blockIdx.x * 8 + wave;
  const int total = E + n;
  if (e >= total) return;
  int s, d;
  if (e < E) { s = ei[e]; d = ei[E + e]; }
  else       { s = e - E; d = s; }
  const float norm = dis[s] * dis[d];
  const float4* fp = (const float4*)(feat + (size_t)s * C) + lane * 2;
  float* op = out + (size_t)d * C + lane * 8;
  float4 v0 = fp[0];
  float4 v1 = fp[1];
  atomicAdd(op + 0, v0.x * norm);
  atomicAdd(op + 1, v0.y * norm);
  atomicAdd(op + 2, v0.z * norm);
  atomicAdd(op + 3, v0.w * norm);
  atomicAdd(op + 4, v1.x * norm);
  atomicAdd(op + 5, v1.y * norm);
  atomicAdd(op + 6, v1.z * norm);
  atomicAdd(op + 7, v1.w * norm);
}

// ---------------- h = relu(h + bias) in place ----------------
__global__ void bias_relu_kernel(float* __restrict__ h, const float* __restrict__ b,
                                 size_t total) {
  size_t i = (size_t)blockIdx.x * blockDim.x + threadIdx.x;
  if (i >= total) return;
  float v = h[i] + b[i & (C - 1)];
  h[i] = v > 0.0f ? v : 0.0f;
}

// ---------------- global mean pool: atomic sums + counts ----------------
__global__ void pool_sum_kernel(const float* __restrict__ h, const int* __restrict__ batch,
                                float* __restrict__ out, float* __restrict__ cnt, int n) {
  int tid = blockIdx.x * blockDim.x + threadIdx.x;
  int node = tid >> 6;          // 64 threads (quads of channels) per node
  int q    = tid & 63;
  if (node >= n) return;
  int g = batch[node];
  const float4 v = *((const float4*)(h + (size_t)node * C) + q);
  float* op = out + (size_t)g * C + q * 4;
  atomicAdd(op + 0, v.x);
  atomicAdd(op + 1, v.y);
  atomicAdd(op + 2, v.z);
  atomicAdd(op + 3, v.w);
  if (q == 0) atomicAdd(&cnt[g], 1.0f);
}

__global__ void pool_div_kernel(float* __restrict__ out, const float* __restrict__ cnt) {
  int i = blockIdx.x * blockDim.x + threadIdx.x;
  if (i >= NUM_GRAPHS * C) return;
  out[i] /= fmaxf(cnt[i >> 8], 1.0f);
}

// ---------------- launch ----------------
extern "C" void kernel_launch(void* const* d_in, const int* in_sizes, int n_in,
                              void* d_out, int out_size, void* d_ws, size_t ws_size,
                              hipStream_t stream) {
  const float* x     = (const float*)d_in[0];
  const int*   ei    = (const int*)  d_in[1];   // [2, E] row-major: src then dst
  const int*   batch = (const int*)  d_in[2];
  const float* W1    = (const float*)d_in[3];
  const float* b1    = (const float*)d_in[4];
  const float* W2    = (const float*)d_in[5];
  const float* b2    = (const float*)d_in[6];
  float* out = (float*)d_out;

  const int n = in_sizes[2];          // 50000 nodes
  const int E = in_sizes[1] / 2;      // 800000 edges
  const size_t nC = (size_t)n * C;

  // workspace layout (floats): [deg/dis : pad(n)] [cnt : 256] [bufA : n*C] [bufB : n*C]
  float* ws   = (float*)d_ws;
  size_t off  = ((size_t)n + 255) & ~(size_t)255;
  float* deg  = ws;                    // becomes deg^{-1/2} in place
  float* cnt  = ws + off;
  float* bufA = cnt + 256;
  float* bufB = bufA + nC;

  const int T = 256;
  const int ET = E + n;

  // degree + normalization
  zero_kernel<<<(int)((off + 256 + T - 1) / T), T, 0, stream>>>(deg, off + 256);
  degree_kernel<<<(ET + T - 1) / T, T, 0, stream>>>(ei, deg, E, n);
  rsqrt_kernel<<<(n + T - 1) / T, T, 0, stream>>>(deg, n);

  dim3 ggrid((n + 127) / 128, C / 64);

  // layer 1
  gemm_kernel<<<ggrid, T, 0, stream>>>(x, W1, bufA, n);
  zero_kernel<<<(int)((nC + T - 1) / T), T, 0, stream>>>(bufB, nC);
  scatter_kernel<<<(ET + 7) / 8, T, 0, stream>>>(ei, deg, bufA, bufB, E, n);
  bias_relu_kernel<<<(int)((nC + T - 1) / T), T, 0, stream>>>(bufB, b1, nC);

  // layer 2 (h1 lives in bufB; GEMM into bufA, then re-zero bufB for aggregation)
  gemm_kernel<<<ggrid, T, 0, stream>>>(bufB, W2, bufA, n);
  zero_kernel<<<(int)((nC + T - 1) / T), T, 0, stream>>>(bufB, nC);
  scatter_kernel<<<(ET + 7) / 8, T, 0, stream>>>(ei, deg, bufA, bufB, E, n);
  bias_relu_kernel<<<(int)((nC + T - 1) / T), T, 0, stream>>>(bufB, b2, nC);

  // global mean pool
  zero_kernel<<<(NUM_GRAPHS * C + T - 1) / T, T, 0, stream>>>(out, (size_t)NUM_GRAPHS * C);
  pool_sum_kernel<<<(int)(((size_t)n * 64 + T - 1) / T), T, 0, stream>>>(bufB, batch, out, cnt, n);
  pool_div_kernel<<<(NUM_GRAPHS * C + T - 1) / T, T, 0, stream>>>(out, cnt);
}